// GATv2Layer_21878563406397
// MI455X (gfx1250) — compile-verified
//
#include <hip/hip_runtime.h>

#define N_NODES 50000
#define N_EDGESC 800000
#define IN_F 128
#define HEADS 4
#define OUT_F 32
#define HF 128          // HEADS*OUT_F
#define NEG_SLOPE 0.2f
#define EPSV 1e-9f

typedef __attribute__((ext_vector_type(2))) float v2f;
typedef __attribute__((ext_vector_type(8))) float v8f;

// ---------------------------------------------------------------------------
// Kernel 1: h = x @ W^T via V_WMMA_F32_16X16X4_F32, register-blocked 4x in N.
// One wave32 owns a 16(node) x 64(outcol) strip: 4 accumulator tiles share
// every A load (x read once per 4 N-tiles instead of once per tile).
// A layout (16x4 f32): lanes 0-15 row M=lane hold {K0,K1}; lanes 16-31 {K2,K3}.
// B layout (4x16 f32): lanes 0-15 col N=lane hold {K0,K1}; lanes 16-31 {K2,K3}.
//   B[k][n] = W[n][k]  (h = x @ W.T), so B column n is W row n.
// D layout: VGPR r -> M = r + 8*(lane>=16), N = lane&15.
// ---------------------------------------------------------------------------
__global__ __launch_bounds__(32) void gatv2_gemm_wmma(
    const float* __restrict__ x, const float* __restrict__ W,
    float* __restrict__ h) {
  const int lane = threadIdx.x;
  const int half = lane >> 4;        // 0: {K0,K1} ; 1: {K2,K3}
  const int l16  = lane & 15;
  const int mt   = blockIdx.x;       // node tile (3125*16 == 50000 exact)
  const int ntb  = blockIdx.y * 4;   // first of 4 out-col tiles

  const int row = mt * 16 + l16;     // A row (node)
  const float* ap = x + (size_t)row * IN_F + 2 * half;
  // B base for tile ntb+j is W row (ntb+j)*16 + l16
  const float* bp = W + ((size_t)(ntb * 16) + l16) * IN_F + 2 * half;

  v8f acc[4] = {};
#pragma unroll 4
  for (int k0 = 0; k0 < IN_F; k0 += 4) {
    v2f a; a.x = ap[k0]; a.y = ap[k0 + 1];   // fuses to global_load_b64
#pragma unroll
    for (int j = 0; j < 4; ++j) {
      const float* bj = bp + (size_t)j * 16 * IN_F;
      v2f b; b.x = bj[k0]; b.y = bj[k0 + 1];
      acc[j] = __builtin_amdgcn_wmma_f32_16x16x4_f32(
          /*neg_a=*/false, a, /*neg_b=*/false, b,
          /*c_mod=*/(short)0, acc[j], /*reuse_a=*/false, /*reuse_b=*/false);
    }
  }

  // Store: row (mt*16 + r + 8*half), col (ntb+j)*16 + l16. Single base addr,
  // all 32 stores use immediate offsets.
  float* hp = h + (size_t)(mt * 16 + 8 * half) * HF + ntb * 16 + l16;
#pragma unroll
  for (int j = 0; j < 4; ++j)
#pragma unroll
    for (int r = 0; r < 8; ++r)
      hp[(size_t)r * HF + j * 16] = acc[j][r];
}

// ---------------------------------------------------------------------------
// Kernel 2: per-(node,head) attention scores on leaky_relu(h).
// ---------------------------------------------------------------------------
__global__ __launch_bounds__(256) void gatv2_scores(
    const float* __restrict__ h, const float* __restrict__ att,
    float* __restrict__ s_src, float* __restrict__ s_dst) {
  const int idx = blockIdx.x * blockDim.x + threadIdx.x;  // node*HEADS + head
  if (idx >= N_NODES * HEADS) return;
  const int n  = idx >> 2;
  const int hd = idx & 3;
  const float* hp = h + (size_t)n * HF + hd * OUT_F;
  const float* a0 = att + hd * (2 * OUT_F);
  const float* a1 = a0 + OUT_F;
  float ss = 0.f, sd = 0.f;
#pragma unroll
  for (int d = 0; d < OUT_F; ++d) {
    const float v  = hp[d];
    const float lv = v > 0.f ? v : v * NEG_SLOPE;
    ss = fmaf(lv, a0[d], ss);
    sd = fmaf(lv, a1[d], sd);
  }
  s_src[idx] = ss;
  s_dst[idx] = sd;
}

// ---------------------------------------------------------------------------
// Kernel 3: out = bias (broadcast), denom = 0.   (harness poisons buffers)
// ---------------------------------------------------------------------------
__global__ __launch_bounds__(128) void gatv2_init(
    float* __restrict__ out, const float* __restrict__ bias,
    float* __restrict__ denom) {
  const int n = blockIdx.x;
  const int f = threadIdx.x;
  out[(size_t)n * HF + f] = bias[f];
  if (f < HEADS) denom[n * HEADS + f] = 0.f;
}

// ---------------------------------------------------------------------------
// Kernel 4: denom[dst,h] += exp(s_src[src,h] + s_dst[dst,h]) over edges
// (self-loops are edges [E, E+N) with src=dst=e-E).  L2-side f32 atomics.
// ---------------------------------------------------------------------------
__global__ __launch_bounds__(256) void gatv2_denom(
    const long long* __restrict__ ei,
    const float* __restrict__ s_src, const float* __restrict__ s_dst,
    float* __restrict__ denom) {
  const int idx = blockIdx.x * blockDim.x + threadIdx.x;
  const int total = (N_EDGESC + N_NODES) * HEADS;
  if (idx >= total) return;
  const int e  = idx >> 2;
  const int hd = idx & 3;
  int src, dst;
  if (e < N_EDGESC) {
    src = (int)ei[e];
    dst = (int)ei[N_EDGESC + e];
  } else {
    src = dst = e - N_EDGESC;
  }
  const float sc = s_src[src * HEADS + hd] + s_dst[dst * HEADS + hd];
  atomicAdd(&denom[dst * HEADS + hd], __expf(sc));
}

// ---------------------------------------------------------------------------
// Kernel 5: out[dst] += h[src] * w  (one wave per edge; lane = feature%32,
// head = feature/32; 4 features per lane; f32 atomics resolved in L2 —
// whole h+out working set (~51 MB) is resident in the 192 MB L2).
// ---------------------------------------------------------------------------
__global__ __launch_bounds__(256) void gatv2_aggregate(
    const long long* __restrict__ ei, const float* __restrict__ h,
    const float* __restrict__ s_src, const float* __restrict__ s_dst,
    const float* __restrict__ denom, float* __restrict__ out) {
  const int lane = threadIdx.x & 31;
  const int e = (blockIdx.x * blockDim.x + threadIdx.x) >> 5;
  if (e >= N_EDGESC + N_NODES) return;
  int src, dst;
  if (e < N_EDGESC) {
    src = (int)ei[e];
    dst = (int)ei[N_EDGESC + e];
  } else {
    src = dst = e - N_EDGESC;
  }
  const float* hs = h + (size_t)src * HF;
  float* op = out + (size_t)dst * HF;
#pragma unroll
  for (int j = 0; j < HEADS; ++j) {
    const float w = __expf(s_src[src * HEADS + j] + s_dst[dst * HEADS + j]) /
                    (denom[dst * HEADS + j] + EPSV);
    const int f = lane + 32 * j;
    atomicAdd(&op[f], hs[f] * w);
  }
}

// ---------------------------------------------------------------------------
// Launch.  Inputs (setup_inputs order): x, edge_index(int64), W, att, bias.
// Workspace floats: h[N*128] | s_src[N*4] | s_dst[N*4] | denom[N*4]  (~28 MB)
// ---------------------------------------------------------------------------
extern "C" void kernel_launch(void* const* d_in, const int* in_sizes, int n_in,
                              void* d_out, int out_size, void* d_ws, size_t ws_size,
                              hipStream_t stream) {
  (void)in_sizes; (void)n_in; (void)out_size; (void)ws_size;
  const float*     x    = (const float*)d_in[0];
  const long long* ei   = (const long long*)d_in[1];
  const float*     W    = (const float*)d_in[2];
  const float*     att  = (const float*)d_in[3];
  const float*     bias = (const float*)d_in[4];
  float* out = (float*)d_out;

  float* ws    = (float*)d_ws;
  float* h     = ws;                                   // N*128
  float* s_src = h + (size_t)N_NODES * HF;             // N*4
  float* s_dst = s_src + (size_t)N_NODES * HEADS;      // N*4
  float* denom = s_dst + (size_t)N_NODES * HEADS;      // N*4

  dim3 g1(N_NODES / 16, 2);                            // 3125 x 2 waves, 4 N-tiles each
  gatv2_gemm_wmma<<<g1, 32, 0, stream>>>(x, W, h);

  gatv2_scores<<<(N_NODES * HEADS + 255) / 256, 256, 0, stream>>>(h, att, s_src, s_dst);

  gatv2_init<<<N_NODES, 128, 0, stream>>>(out, bias, denom);

  const int etot = N_EDGESC + N_NODES;
  gatv2_denom<<<(etot * HEADS + 255) / 256, 256, 0, stream>>>(ei, s_src, s_dst, denom);

  gatv2_aggregate<<<((size_t)etot * 32 + 255) / 256, 256, 0, stream>>>(
      ei, h, s_src, s_dst, denom, out);
}